// SpectralMixer_2D_28381143892313
// MI455X (gfx1250) — compile-verified
//
#include <hip/hip_runtime.h>

typedef __attribute__((ext_vector_type(2))) float v2f;
typedef __attribute__((ext_vector_type(4))) float v4f;
typedef __attribute__((ext_vector_type(8))) float v8f;

namespace {
constexpr int B = 8, C = 64, N1 = 256, N2 = 256, MODE = 128;
constexpr int K1 = 96;          // selected block size (2*48)
constexpr int H = 48;           // half modes
constexpr int CHUNK = 32;       // K tile
constexpr int LDA = 34;         // padded A-plane row stride (32 data + 2 pad, keeps 8B align)
constexpr int LDB = 96;         // B-plane row stride
constexpr int NT = K1 / 16;     // 6 tile rows/cols; block = 6 waves, wave w = tile row w
constexpr int THREADS = 32 * NT;  // 192

// index maps: selected rows/cols of x (N=256) and W (MODE=128)
__device__ __forceinline__ int map_x(int i) { return i < H ? i : i + (N1 - K1); }   // +160
__device__ __forceinline__ int map_w(int i) { return i < H ? i : i + (MODE - K1); } // +32
} // namespace

// ---------------------------------------------------------------------------
// Kernel 1: bulk copy x -> out (268 MB each way), nontemporal to spare L2.
// ---------------------------------------------------------------------------
__global__ void copy_x_kernel(const v4f* __restrict__ src,
                              v4f* __restrict__ dst, int n) {
  int i = blockIdx.x * blockDim.x + threadIdx.x;
  const int stride = gridDim.x * blockDim.x;
  for (; i < n; i += stride) {
    v4f v = __builtin_nontemporal_load(src + i);
    __builtin_nontemporal_store(v, dst + i);
  }
}

// ---------------------------------------------------------------------------
// Kernel 2/3: one (b,c) pair per workgroup; complex 96x96x96 GEMM via
// V_WMMA_F32_16X16X4_F32. 6 waves; wave w computes output tile row w (6 tiles).
//   PHASE 1: T1 = W1sel @ Xsel          (A = W1sel,  B = Xsel,  out -> corner)
//   PHASE 2: Out = T1 @ W2sel^T         (A = T1 from corner, B = W2sel^T)
// ---------------------------------------------------------------------------
template <int PHASE>
__global__ __launch_bounds__(THREADS) void spectral_gemm_kernel(
    const float2* __restrict__ x, const float2* __restrict__ W,
    float2* __restrict__ out) {
  __shared__ float Ar[K1 * LDA];
  __shared__ float Ai[K1 * LDA];
  __shared__ float Br[CHUNK * LDB];
  __shared__ float Bi[CHUNK * LDB];

  const int bc = blockIdx.x;       // b*C + c
  const int c = bc & (C - 1);
  const int tid = threadIdx.x;
  const int wave = tid >> 5;       // == tile row tm
  const int lane = tid & 31;

  const float2* __restrict__ Wc = W + (size_t)c * MODE * MODE;
  const float2* __restrict__ Xp = x + (size_t)bc * N1 * N2;
  float2* __restrict__ Op = out + (size_t)bc * N1 * N2;

  const v8f vzero = {0.f, 0.f, 0.f, 0.f, 0.f, 0.f, 0.f, 0.f};
  v8f accR[NT], accI[NT];
#pragma unroll
  for (int i = 0; i < NT; ++i) { accR[i] = vzero; accI[i] = vzero; }

  const int mrow = lane & 15;        // M (or N) within 16
  const int kof = (lane >> 4) << 1;  // K pair select per f32 A/B layout

  for (int ch = 0; ch < 3; ++ch) {
    const int j0 = ch * CHUNK;

    // --- stage A tile (96 x 32 complex, split planes) ---
    for (int idx = tid; idx < K1 * CHUNK; idx += THREADS) {
      const int i = idx >> 5;
      const int jj = idx & 31;
      float2 v;
      if (PHASE == 1) v = Wc[map_w(i) * MODE + map_w(j0 + jj)];
      else            v = Op[map_x(i) * N2 + map_x(j0 + jj)];  // staged T1
      Ar[i * LDA + jj] = v.x;
      Ai[i * LDA + jj] = v.y;
    }
    // --- stage B tile (32 x 96 complex, split planes); phase2 transposes W2 ---
    for (int idx = tid; idx < CHUNK * K1; idx += THREADS) {
      int r, cl;
      float2 v;
      if (PHASE == 1) {
        r = idx / K1; cl = idx - r * K1;                 // B[j][k] = X[rx(j)][cx(k)]
        v = Xp[map_x(j0 + r) * N2 + map_x(cl)];
      } else {
        cl = idx >> 5; r = idx & 31;                     // B[k][l] = W2[rw(l)][cw(k)]
        v = Wc[map_w(cl) * MODE + map_w(j0 + r)];
      }
      Br[r * LDB + cl] = v.x;
      Bi[r * LDB + cl] = v.y;
    }
    __syncthreads();

#pragma unroll
    for (int kk = 0; kk < CHUNK; kk += 4) {
      // A fragment fixed per wave (tm == wave): load once, reuse for 6 tiles
      const int aoff = (wave * 16 + mrow) * LDA + kk + kof;
      const v2f a_r = *(const v2f*)&Ar[aoff];
      const v2f a_i = *(const v2f*)&Ai[aoff];
#pragma unroll
      for (int tn = 0; tn < NT; ++tn) {
        const int boff = (kk + kof) * LDB + tn * 16 + mrow;
        v2f b_r, b_i;
        b_r.x = Br[boff]; b_r.y = Br[boff + LDB];
        b_i.x = Bi[boff]; b_i.y = Bi[boff + LDB];
        const v2f b_in = -b_i;
        // Re += Ar*Br - Ai*Bi ; Im += Ar*Bi + Ai*Br
        accR[tn] = __builtin_amdgcn_wmma_f32_16x16x4_f32(
            false, a_r, false, b_r, (short)0, accR[tn], false, false);
        accR[tn] = __builtin_amdgcn_wmma_f32_16x16x4_f32(
            false, a_i, false, b_in, (short)0, accR[tn], false, false);
        accI[tn] = __builtin_amdgcn_wmma_f32_16x16x4_f32(
            false, a_r, false, b_i, (short)0, accI[tn], false, false);
        accI[tn] = __builtin_amdgcn_wmma_f32_16x16x4_f32(
            false, a_i, false, b_r, (short)0, accI[tn], false, false);
      }
    }
    __syncthreads();
  }

  // --- store D fragments to the scattered corner of out ---
  const int mo = (lane >> 4) * 8;  // lanes 16..31 hold M = j + 8
  const int n = lane & 15;
#pragma unroll
  for (int tn = 0; tn < NT; ++tn) {
    const int col = map_x(tn * 16 + n);
#pragma unroll
    for (int j = 0; j < 8; ++j) {
      const int row = map_x(wave * 16 + mo + j);
      float2 o;
      o.x = accR[tn][j];
      o.y = accI[tn][j];
      Op[row * N2 + col] = o;
    }
  }
}

// ---------------------------------------------------------------------------
extern "C" void kernel_launch(void* const* d_in, const int* in_sizes, int n_in,
                              void* d_out, int out_size, void* d_ws,
                              size_t ws_size, hipStream_t stream) {
  (void)in_sizes; (void)n_in; (void)d_ws; (void)ws_size; (void)out_size;
  const float2* x = (const float2*)d_in[0];    // complex64 interleaved
  const float2* W1 = (const float2*)d_in[1];
  const float2* W2 = (const float2*)d_in[2];
  float2* out = (float2*)d_out;

  const int n4 = (B * C * N1 * N2) / 2;  // float4 count (2 complex each)
  copy_x_kernel<<<8192, 256, 0, stream>>>((const v4f*)x, (v4f*)out, n4);
  spectral_gemm_kernel<1><<<B * C, THREADS, 0, stream>>>(x, W1, out);
  spectral_gemm_kernel<2><<<B * C, THREADS, 0, stream>>>(x, W2, out);
}